// FocalTverskyLoss_21749714387529
// MI455X (gfx1250) — compile-verified
//
#include <hip/hip_runtime.h>
#include <stdint.h>

// ---------------------------------------------------------------------------
// Focal Tversky loss, B=8 C=21 H=W=512, f32.
// Memory-bound: 176MB inputs + 8MB targets, one pass -> ~7.9us @ 23.3TB/s.
// CDNA5 path: Tensor Data Mover streams strided [21 x 1024]-f32 tiles
// (rows 1MB apart in the NCHW layout) into LDS, double-buffered, synced with
// TENSORcnt; compute reads ds_load_b128 and does an online-stable softmax.
// No matmul structure -> WMMA intentionally unused (would not reduce bytes).
// ---------------------------------------------------------------------------

static constexpr int B = 8, C = 21, H = 512, W = 512;
static constexpr int HW = H * W;                       // 262144
static constexpr unsigned IGNORE_INDEX = 255u;
static constexpr float ALPHA = 0.7f, BETA = 0.3f, SMOOTH = 1.0f;
static constexpr float GAMMA_E = 4.0f / 3.0f;

static constexpr int THREADS = 256;                    // 8 x wave32
static constexpr int CHUNK = 1024;                     // pixels per TDM tile
static constexpr int PIXPT = CHUNK / THREADS;          // 4 pixels / thread
static constexpr int BLOCKS_PER_BATCH = 64;
static constexpr int CHUNKS_PER_BLOCK = HW / (BLOCKS_PER_BATCH * CHUNK);  // 4
static constexpr int GRID = B * BLOCKS_PER_BATCH;      // 512 blocks
static constexpr size_t SMEM_BYTES = 2ull * C * CHUNK * sizeof(float);    // 168KB

typedef unsigned int u32x4 __attribute__((ext_vector_type(4)));
typedef int          i32x4 __attribute__((ext_vector_type(4)));
typedef int          i32x8 __attribute__((ext_vector_type(8)));

#if defined(__HIP_DEVICE_COMPILE__) && __has_builtin(__builtin_amdgcn_tensor_load_to_lds)
#define HAVE_TDM 1
#else
#define HAVE_TDM 0
#endif

#if HAVE_TDM
// Issue one TDM DMA: 2D tile, tile_dim0=CHUNK floats (contiguous),
// tile_dim1=C rows, row stride = HW floats. Written densely into LDS at ldst.
__device__ __forceinline__ void tdm_issue(const float* gsrc, float* ldst) {
  const uint64_t ga = (uint64_t)(uintptr_t)gsrc;
  const uint32_t la = (uint32_t)(uintptr_t)ldst;   // low 32b of flat = LDS byte offset
  u32x4 g0;
  g0.x = 1u;                                       // count=1 (valid), user mode
  g0.y = la;                                       // lds_addr
  g0.z = (uint32_t)ga;                             // global_addr[31:0]
  g0.w = (uint32_t)((ga >> 32) & 0x1FFFFFFu)       // global_addr[56:32]
       | (2u << 30);                               // type=2 ("image")
  i32x8 g1;
  g1[0] = (2 << 16);                               // data_size=4B; no pad/iter/multicast
  g1[1] = (int)((CHUNK & 0xFFFF) << 16);           // tensor_dim0[15:0] = CHUNK
  g1[2] = (int)(C << 16);                          // tensor_dim0 hi=0 | tensor_dim1 lo=C
  g1[3] = (int)((CHUNK & 0xFFFF) << 16);           // tensor_dim1 hi=0 | tile_dim0=CHUNK
  g1[4] = C;                                       // tile_dim1=C, tile_dim2=0 (2D)
  g1[5] = HW;                                      // tensor_dim0_stride lo
  g1[6] = 0;                                       // stride hi=0 | dim1_stride lo (unused)
  g1[7] = 0;
  const i32x4 z4 = {0, 0, 0, 0};
#if __clang_major__ >= 23
  const i32x8 z8 = {0, 0, 0, 0, 0, 0, 0, 0};
  __builtin_amdgcn_tensor_load_to_lds(g0, g1, z4, z4, z8, 0);
#else
  __builtin_amdgcn_tensor_load_to_lds(g0, g1, z4, z4, 0);
#endif
}
#endif  // HAVE_TDM

__global__ __launch_bounds__(THREADS, 1) void ftl_main(
    const float* __restrict__ inputs, const int* __restrict__ targets,
    float* __restrict__ ws) {
  extern __shared__ float smem[];
  const int tid = threadIdx.x;
  const int b   = blockIdx.x / BLOCKS_PER_BATCH;
  const int blk = blockIdx.x % BLOCKS_PER_BATCH;
  const int base_pix = blk * (CHUNKS_PER_BLOCK * CHUNK);
  const float* in_b = inputs + (size_t)b * C * HW;
  const int*   tg_b = targets + (size_t)b * HW;

  float tp = 0.0f, cnt = 0.0f;

#if HAVE_TDM
  if (tid == 0) tdm_issue(in_b + base_pix, &smem[0]);   // prefetch chunk 0
#endif

  for (int ch = 0; ch < CHUNKS_PER_BLOCK; ++ch) {
    const int cs = base_pix + ch * CHUNK;
#if HAVE_TDM
    float* buf = &smem[(ch & 1) * (C * CHUNK)];
    // Prefetch next chunk into the other buffer while we compute this one.
    if (tid == 0 && (ch + 1 < CHUNKS_PER_BLOCK))
      tdm_issue(in_b + cs + CHUNK, &smem[((ch + 1) & 1) * (C * CHUNK)]);
    if (tid < 32) {  // wave 0 owns TENSORcnt (in-order completion)
      if (ch + 1 < CHUNKS_PER_BLOCK) __builtin_amdgcn_s_wait_tensorcnt(1);
      else                           __builtin_amdgcn_s_wait_tensorcnt(0);
    }
    __syncthreads();                                    // publish tile to all waves
    const float* src = buf;
    const int sstride = CHUNK;
#else
    const float* src = in_b + cs;                       // direct coalesced b128 loads
    const int sstride = HW;
#endif
    const int p0 = tid * PIXPT;
    const int4 tq = *(const int4*)&tg_b[cs + p0];
    const int tg4[4] = {tq.x, tq.y, tq.z, tq.w};
    float m[4], s[4], vt[4];
#pragma unroll
    for (int k = 0; k < 4; ++k) { m[k] = -3.0e38f; s[k] = 0.0f; vt[k] = 0.0f; }
    // Online-stable softmax accumulation over the 21 classes.
#pragma unroll
    for (int c = 0; c < C; ++c) {
      const float4 v = *(const float4*)&src[c * sstride + p0];
      const float vv[4] = {v.x, v.y, v.z, v.w};
#pragma unroll
      for (int k = 0; k < 4; ++k) {
        const float x  = vv[k];
        const float mn = fmaxf(m[k], x);
        s[k] = s[k] * __expf(m[k] - mn) + __expf(x - mn);
        m[k] = mn;
        if (c == tg4[k]) vt[k] = x;
      }
    }
#pragma unroll
    for (int k = 0; k < 4; ++k) {
      if ((unsigned)tg4[k] != IGNORE_INDEX) {
        tp  += __expf(vt[k] - m[k]) / s[k];   // p_t for the true class
        cnt += 1.0f;
      }
    }
#if HAVE_TDM
    __syncthreads();   // everyone done reading buf before DMA reuses it
#endif
  }

  // Deterministic block reduction: wave32 shuffles -> LDS -> per-block partial.
#pragma unroll
  for (int off = 16; off > 0; off >>= 1) {
    tp  += __shfl_xor(tp,  off, 32);
    cnt += __shfl_xor(cnt, off, 32);
  }
  __shared__ float r_tp[THREADS / 32], r_cnt[THREADS / 32];
  const int wid = tid >> 5;
  if ((tid & 31) == 0) { r_tp[wid] = tp; r_cnt[wid] = cnt; }
  __syncthreads();
  if (tid == 0) {
    float T = 0.0f, N = 0.0f;
#pragma unroll
    for (int i = 0; i < THREADS / 32; ++i) { T += r_tp[i]; N += r_cnt[i]; }
    ws[blockIdx.x]        = T;   // TP partial
    ws[GRID + blockIdx.x] = N;   // valid-count partial
  }
}

// Fold the 512 partials (64 per batch = 2 waves of 32 threads each),
// apply focal Tversky per batch, mean over batches. Fully deterministic.
__global__ void ftl_final(const float* __restrict__ ws, float* __restrict__ out) {
  __shared__ float s_tp[16], s_cnt[16];
  const int tid = threadIdx.x;            // 512 threads = 16 wave32
  float tp = ws[tid];
  float cn = ws[GRID + tid];
#pragma unroll
  for (int off = 16; off > 0; off >>= 1) {
    tp += __shfl_xor(tp, off, 32);
    cn += __shfl_xor(cn, off, 32);
  }
  const int wid = tid >> 5;
  if ((tid & 31) == 0) { s_tp[wid] = tp; s_cnt[wid] = cn; }
  __syncthreads();
  if (tid == 0) {
    float acc = 0.0f;
#pragma unroll
    for (int b = 0; b < B; ++b) {
      const float TP = s_tp[2 * b] + s_tp[2 * b + 1];
      const float n  = s_cnt[2 * b] + s_cnt[2 * b + 1];
      const float FPFN = n - TP;          // FP == FN == n_valid - TP
      const float tv = (TP + SMOOTH) / (TP + ALPHA * FPFN + BETA * FPFN + SMOOTH);
      const float x  = 1.0f - tv;
      acc += powf(x, GAMMA_E);
    }
    out[0] = acc * (1.0f / (float)B);
  }
}

extern "C" void kernel_launch(void* const* d_in, const int* in_sizes, int n_in,
                              void* d_out, int out_size, void* d_ws, size_t ws_size,
                              hipStream_t stream) {
  (void)in_sizes; (void)n_in; (void)out_size; (void)ws_size;
  const float* inputs  = (const float*)d_in[0];
  const int*   targets = (const int*)d_in[1];
  float* ws  = (float*)d_ws;
  float* out = (float*)d_out;

  // Allow 168KB of dynamic LDS (gfx1250 WGP has 320KB). Deterministic, not a
  // stream op, safe under graph capture.
  hipFuncSetAttribute(reinterpret_cast<const void*>(&ftl_main),
                      hipFuncAttributeMaxDynamicSharedMemorySize,
                      (int)SMEM_BYTES);

  ftl_main<<<GRID, THREADS, SMEM_BYTES, stream>>>(inputs, targets, ws);
  ftl_final<<<1, GRID, 0, stream>>>(ws, out);   // GRID==512 threads, 1 block
}